// GNNConv_61993557950862
// MI455X (gfx1250) — compile-verified
//
#include <hip/hip_runtime.h>
#include <stdint.h>

// LightGCN scatter-add: out[dst[e]] += w[e] * x[src[e]], d=64 f32 features.
// Roofline: 160 MFLOP (irrelevant); ~76 MB mandatory HBM traffic (~3.3us at
// 23.3 TB/s); true bound is E*64 = 80M f32 L2 atomics (unsorted scatter).
// Edge records stream global->LDS via gfx1250 async-to-LDS DMA (ASYNCcnt),
// double buffered so DMA overlaps the gather+atomic phase.

#define TILE_E    128          // edges per LDS tile
#define THREADS   256          // 8 wave32
#define BUF_BYTES 2560         // 1024 (src i64) + 1024 (dst i64) + 512 (w f32)

__device__ __forceinline__ void async_copy_b128(uint32_t lds_off, const void* gptr) {
  // gfx1250: copies 16 bytes from global address vaddr to LDS offset vdst, tracked by ASYNCcnt.
  asm volatile("global_load_async_to_lds_b128 %0, %1, off"
               :: "v"(lds_off), "v"(gptr)
               : "memory");
}

__device__ __forceinline__ void stage_tile(const char* src8, const char* dst8,
                                           const char* w4, long long base, int nE,
                                           char* buf) {
  // 160 x 16B chunks: threads 0..63 src, 64..127 dst, 128..159 w.
  // Branch-free value selection -> single predicated async issue
  // (exactly one ASYNCcnt tick per wave 0..4 per tile).
  const int tid = (int)threadIdx.x;
  if (tid < 160) {
    const uint32_t lds = (uint32_t)(uintptr_t)buf;  // generic LDS ptr: low 32b = offset
    const bool isSrc = tid < 64;
    const bool isDst = !isSrc && tid < 128;
    const int  lane  = isSrc ? tid : (isDst ? tid - 64 : tid - 128);
    const int  off   = lane << 4;
    const uint32_t ldsOff = lds + (isSrc ? 0u : (isDst ? 1024u : 2048u)) + off;
    const char* g = (isSrc ? src8 + base * 8
                           : (isDst ? dst8 + base * 8 : w4 + base * 4)) + off;
    const int lim = (isSrc | isDst) ? nE * 8 : nE * 4;
    if (off < lim) async_copy_b128(ldsOff, g);
  }
}

__global__ __launch_bounds__(THREADS)
void lightgcn_scatter(const float* __restrict__ x,
                      const long long* __restrict__ eidx,  // [2, E]
                      const float* __restrict__ w,
                      float* __restrict__ out,
                      int E, int nTiles) {
  __shared__ __align__(16) char sh[2 * BUF_BYTES];
  const char* src8 = (const char*)eidx;        // edge_index[0, :]
  const char* dst8 = (const char*)(eidx + E);  // edge_index[1, :]
  const char* w4   = (const char*)w;

  const int tid = (int)threadIdx.x;
  const int t0  = (int)blockIdx.x;
  if (t0 >= nTiles) return;
  const int stride = (int)gridDim.x;

  const int g = tid >> 4;        // edge slot within sweep (0..15)
  const int q = (tid & 15) * 4;  // float offset of this lane's quad in the row

  // Prologue: stage first tile into buffer 0.
  {
    const long long base = (long long)t0 * TILE_E;
    const int nE = min(TILE_E, E - (int)base);
    stage_tile(src8, dst8, w4, base, nE, sh);
  }

  int parity = 0;
  for (int t = t0; t < nTiles; t += stride, parity ^= 1) {
    const int tn = t + stride;
    if (tn < nTiles) {
      // Kick off DMA for the next tile into the other buffer, then wait for
      // the *current* tile only (per-wave in-order ASYNCcnt).
      const long long baseN = (long long)tn * TILE_E;
      const int nEn = min(TILE_E, E - (int)baseN);
      stage_tile(src8, dst8, w4, baseN, nEn, sh + (parity ? 0 : BUF_BYTES));
      asm volatile("s_wait_asynccnt 1" ::: "memory");
    } else {
      asm volatile("s_wait_asynccnt 0" ::: "memory");
    }
    __syncthreads();  // staged edge records visible to all 8 waves

    const char*     buf = sh + (parity ? BUF_BYTES : 0);
    const uint32_t* sl  = (const uint32_t*)buf;            // int64 src, low dword
    const uint32_t* dl  = (const uint32_t*)(buf + 1024);   // int64 dst, low dword
    const float*    wl  = (const float*)(buf + 2048);

    const int nE = min(TILE_E, E - t * TILE_E);

    if (nE == TILE_E) {
      // Fast path: full tile, no per-sweep guards in the hot loop.
#pragma unroll
      for (int i = 0; i < TILE_E / 16; ++i) {
        const int e = i * 16 + g;
        const uint32_t s  = sl[2 * e];
        const uint32_t d  = dl[2 * e];
        const float    ww = wl[e];
        // 16 lanes x 16B = one coalesced 256B row gather per edge.
        const float4 xv = *(const float4*)(x + (size_t)s * 64 + q);
        float* o = out + (size_t)d * 64 + q;
        __hip_atomic_fetch_add(o + 0, ww * xv.x, __ATOMIC_RELAXED, __HIP_MEMORY_SCOPE_AGENT);
        __hip_atomic_fetch_add(o + 1, ww * xv.y, __ATOMIC_RELAXED, __HIP_MEMORY_SCOPE_AGENT);
        __hip_atomic_fetch_add(o + 2, ww * xv.z, __ATOMIC_RELAXED, __HIP_MEMORY_SCOPE_AGENT);
        __hip_atomic_fetch_add(o + 3, ww * xv.w, __ATOMIC_RELAXED, __HIP_MEMORY_SCOPE_AGENT);
      }
    } else {
      // Tail tile (once per block at most).
#pragma unroll
      for (int i = 0; i < TILE_E / 16; ++i) {
        const int e = i * 16 + g;
        if (e < nE) {
          const uint32_t s  = sl[2 * e];
          const uint32_t d  = dl[2 * e];
          const float    ww = wl[e];
          const float4 xv = *(const float4*)(x + (size_t)s * 64 + q);
          float* o = out + (size_t)d * 64 + q;
          __hip_atomic_fetch_add(o + 0, ww * xv.x, __ATOMIC_RELAXED, __HIP_MEMORY_SCOPE_AGENT);
          __hip_atomic_fetch_add(o + 1, ww * xv.y, __ATOMIC_RELAXED, __HIP_MEMORY_SCOPE_AGENT);
          __hip_atomic_fetch_add(o + 2, ww * xv.z, __ATOMIC_RELAXED, __HIP_MEMORY_SCOPE_AGENT);
          __hip_atomic_fetch_add(o + 3, ww * xv.w, __ATOMIC_RELAXED, __HIP_MEMORY_SCOPE_AGENT);
        }
      }
    }
    __syncthreads();  // buffer reuse: all waves done reading before next DMA targets it
  }
}

extern "C" void kernel_launch(void* const* d_in, const int* in_sizes, int n_in,
                              void* d_out, int out_size, void* d_ws, size_t ws_size,
                              hipStream_t stream) {
  (void)n_in; (void)d_ws; (void)ws_size;
  const float*     x    = (const float*)d_in[0];
  const long long* eidx = (const long long*)d_in[1];   // int64 [2, E]
  const float*     w    = (const float*)d_in[2];
  float*           out  = (float*)d_out;
  const int E = in_sizes[2];                           // 1,250,000

  // Harness poisons d_out; zero it (memset nodes are graph-capturable).
  hipMemsetAsync(d_out, 0, (size_t)out_size * sizeof(float), stream);

  const int nTiles  = (E + TILE_E - 1) / TILE_E;
  const int nBlocks = nTiles < 2048 ? nTiles : 2048;   // max outstanding atomics
  lightgcn_scatter<<<nBlocks, THREADS, 0, stream>>>(x, eidx, w, out, E, nTiles);
}